// Ortho_GCN_60498909331914
// MI455X (gfx1250) — compile-verified
//
#include <hip/hip_runtime.h>
#include <math.h>

typedef __attribute__((ext_vector_type(2))) float v2f;
typedef __attribute__((ext_vector_type(8))) float v8f;

#define NDIM 16384
#define DDIM 128
#define EPS  1e-5f

// ---------------------------------------------------------------------------
// Kernel 1: ortho_trans(weight) -> t   (tiny; one block per group, VALU+LDS)
// ---------------------------------------------------------------------------
__device__ __forceinline__ void mm64(float* __restrict__ Cm,
                                     const float* __restrict__ Am,
                                     const float* __restrict__ Bmat,
                                     int tid) {
  // Cm = Am @ Bmat, all 64x64 row-major in LDS
  for (int idx = tid; idx < 64 * 64; idx += 256) {
    int i = idx >> 6, j = idx & 63;
    float s = 0.0f;
#pragma unroll 8
    for (int k = 0; k < 64; ++k)
      s = fmaf(Am[(i << 6) + k], Bmat[(k << 6) + j], s);
    Cm[idx] = s;
  }
}

__global__ void __launch_bounds__(256)
ortho_kernel(const float* __restrict__ weight, float* __restrict__ tOut) {
  __shared__ float S[64 * 64];
  __shared__ float Bm[64 * 64];
  __shared__ float T1[64 * 64];
  __shared__ float T2[64 * 64];

  const int g   = blockIdx.x;            // group 0 or 1
  const int tid = threadIdx.x;
  const float* W = weight + g * 64 * DDIM;   // this group's 64x128 slice

  // row means -> T1[0..63]
  for (int r = tid; r < 64; r += 256) {
    float s = 0.0f;
    for (int k = 0; k < DDIM; ++k) s += W[r * DDIM + k];
    T1[r] = s * (1.0f / DDIM);
  }
  __syncthreads();

  // S = Zc @ Zc^T + EPS*I   (Zc recomputed on the fly)
  for (int idx = tid; idx < 64 * 64; idx += 256) {
    int i = idx >> 6, j = idx & 63;
    float mi = T1[i], mj = T1[j];
    float s = 0.0f;
    for (int k = 0; k < DDIM; ++k)
      s = fmaf(W[i * DDIM + k] - mi, W[j * DDIM + k] - mj, s);
    if (i == j) s += EPS;
    S[idx] = s;
  }
  __syncthreads();

  // Frobenius norm of S (reduce through T2 scratch)
  float p = 0.0f;
  for (int idx = tid; idx < 64 * 64; idx += 256) p = fmaf(S[idx], S[idx], p);
  T2[tid] = p;
  __syncthreads();
  for (int s = 128; s > 0; s >>= 1) {
    if (tid < s) T2[tid] += T2[tid + s];
    __syncthreads();
  }
  const float normS = sqrtf(T2[0]);
  __syncthreads();

  // S /= normS ; Bm = I
  const float rn = 1.0f / normS;
  for (int idx = tid; idx < 64 * 64; idx += 256) {
    S[idx] *= rn;
    Bm[idx] = ((idx >> 6) == (idx & 63)) ? 1.0f : 0.0f;
  }
  __syncthreads();

  // Newton-Schulz: B <- 1.5B - 0.5 * B*(B*(B*S))
  for (int it = 0; it < 4; ++it) {
    mm64(T1, Bm, S, tid);  __syncthreads();   // T1 = B  S
    mm64(T2, Bm, T1, tid); __syncthreads();   // T2 = B2 S
    mm64(T1, Bm, T2, tid); __syncthreads();   // T1 = B3 S
    for (int idx = tid; idx < 64 * 64; idx += 256)
      Bm[idx] = 1.5f * Bm[idx] - 0.5f * T1[idx];
    __syncthreads();
  }

  // recompute row means into T2[0..63]
  for (int r = tid; r < 64; r += 256) {
    float s = 0.0f;
    for (int k = 0; k < DDIM; ++k) s += W[r * DDIM + k];
    T2[r] = s * (1.0f / DDIM);
  }
  __syncthreads();

  // t[g*64+i][k] = (Bm @ Zc)[i][k] / sqrt(normS)
  const float rs = rsqrtf(normS);
  for (int idx = tid; idx < 64 * DDIM; idx += 256) {
    int i = idx >> 7, k = idx & 127;
    float s = 0.0f;
#pragma unroll 8
    for (int j = 0; j < 64; ++j)
      s = fmaf(Bm[(i << 6) + j], W[j * DDIM + k] - T2[j], s);
    tOut[(g * 64 + i) * DDIM + k] = s * rs;
  }
}

// ---------------------------------------------------------------------------
// Kernel 2/3: C[M x 128] = (relu?) A[M x K] @ B[K x 128]  via WMMA f32 16x16x4
// One wave per 16x128 output slab; 8 waves (256 thr) per block => 128 rows.
// ---------------------------------------------------------------------------
template <bool RELU>
__global__ void __launch_bounds__(256)
gemm128_wmma(const float* __restrict__ A, const float* __restrict__ B,
             float* __restrict__ C, int M, int K) {
  const int lane  = threadIdx.x & 31;
  const int wave  = threadIdx.x >> 5;
  const int row0  = (blockIdx.x * 8 + wave) << 4;   // 16-row tile origin
  const int mrow  = lane & 15;                      // M index inside tile
  const int khalf = lane >> 4;                      // 0 -> K0/K1, 1 -> K2/K3
  const int nlane = lane & 15;                      // N index inside tile

  // A fragment stream: lane reads float2 {k, k+1} (or {k+2, k+3}) of its row
  const float* arow = A + (size_t)(row0 + mrow) * (size_t)K + (khalf << 1);
  // B fragment base: lane reads rows (k + 2*khalf) and (k + 2*khalf + 1), col nlane
  const float* bcol = B + nlane + (khalf << 8);     // khalf*2 rows * 128

  v8f acc[8] = {};

  for (int k0 = 0; k0 < K; k0 += 4) {
    v2f a = *(const v2f*)(arow + k0);
    __builtin_prefetch(arow + k0 + 128, 0, 0);      // ~512B ahead on the g stream
    const float* bk = bcol + (size_t)k0 * DDIM;
#pragma unroll
    for (int t = 0; t < 8; ++t) {
      v2f b;
      b.x = bk[t * 16];
      b.y = bk[t * 16 + DDIM];
      acc[t] = __builtin_amdgcn_wmma_f32_16x16x4_f32(
          /*neg_a=*/false, a, /*neg_b=*/false, b,
          /*c_mod=*/(short)0, acc[t], /*reuse_a=*/false, /*reuse_b=*/false);
    }
  }

  // C/D layout: VGPR v -> row (v + 8*khalf), lane -> col
  const int rbase = row0 + (khalf << 3);
#pragma unroll
  for (int t = 0; t < 8; ++t) {
#pragma unroll
    for (int v = 0; v < 8; ++v) {
      float val = acc[t][v];
      if (RELU) val = fmaxf(val, 0.0f);
      C[(size_t)(rbase + v) * DDIM + t * 16 + nlane] = val;
    }
  }
}

// ---------------------------------------------------------------------------
extern "C" void kernel_launch(void* const* d_in, const int* in_sizes, int n_in,
                              void* d_out, int out_size, void* d_ws, size_t ws_size,
                              hipStream_t stream) {
  const float* g = (const float*)d_in[0];   // [N, N]
  const float* h = (const float*)d_in[1];   // [N, D]
  const float* w = (const float*)d_in[2];   // [D, D]
  float* out  = (float*)d_out;              // [N, D]

  float* tbuf = (float*)d_ws;               // [D, D]   orthogonalized weight
  float* mbuf = tbuf + DDIM * DDIM;         // [N, D]   m = h @ t

  // 1) t = ortho_trans(weight)
  ortho_kernel<<<2, 256, 0, stream>>>(w, tbuf);

  // 2) m = h @ t       (M=16384, K=128)
  gemm128_wmma<false><<<NDIM / 128, 256, 0, stream>>>(h, tbuf, mbuf, NDIM, DDIM);

  // 3) out = relu(g @ m)  (M=16384, K=16384; streams g once from HBM)
  gemm128_wmma<true><<<NDIM / 128, 256, 0, stream>>>(g, mbuf, out, NDIM, NDIM);
}